// OptLayer_74672301408263
// MI455X (gfx1250) — compile-verified
//
#include <hip/hip_runtime.h>

// ---------------------------------------------------------------------------
// OptLayer IPM-QP solver for MI455X (gfx1250, wave32).
// One wave per batch sample.
//  - KKT build:  M = Q + Ss*diag(w_top)*Ss^T + diag(w_bot) + eps*I
//                via V_WMMA_F32_16X16X4_F32 from LDS-resident Ss^T and Q.
//  - Factor:     blocked Cholesky (NB=16); panel scalar, trailing SYRK
//                (A22 -= P P^T) on WMMA with negated A-fragments.
//  - Solves:     wave-cooperative forward/backward substitution in LDS.
// ---------------------------------------------------------------------------

typedef float v2f __attribute__((ext_vector_type(2)));
typedef float v8f __attribute__((ext_vector_type(8)));

#define NRATES   64
#define NSPEC    32
#define NCON     96
#define NITER    20
#define REG_JIT  1e-6f
#define SIGMA    0.1f
#define TAU      0.995f

#define SST_STR  65   // Ss^T rows: 32 x 65 (padded; +1 -> conflict-free columns)
#define Q_STR    66   // Q rows: 64 x 66 (padded; +2 -> disjoint banks per C half)
#define M_STR    66   // M rows: 64 x 66 (stride 66 = 2 mod 64 -> conflict-free cols)
#define WPG      2    // waves (samples) per workgroup

// per-wave LDS fence: order cross-lane LDS write -> read phases
#define LDS_SYNC() asm volatile("s_wait_dscnt 0" ::: "memory")

__device__ __forceinline__ float wave_sum32(float v) {
#pragma unroll
    for (int off = 16; off > 0; off >>= 1) v += __shfl_xor(v, off, 32);
    return v;
}
__device__ __forceinline__ float wave_min32(float v) {
#pragma unroll
    for (int off = 16; off > 0; off >>= 1) v = fminf(v, __shfl_xor(v, off, 32));
    return v;
}

__global__ __launch_bounds__(32 * WPG)
void optlayer_ipm_kernel(const float* __restrict__ dc_in,
                         const float* __restrict__ c_in,
                         const float* __restrict__ SsG,
                         const float* __restrict__ Qg,
                         const float* __restrict__ maxc,
                         float* __restrict__ out)
{
    __shared__ float lds_SsT[NSPEC  * SST_STR];     //  8320 B
    __shared__ float lds_Q  [NRATES * Q_STR];       // 16896 B
    __shared__ float lds_M  [WPG][NRATES * M_STR];  // 33792 B
    __shared__ float lds_vec[WPG][NCON];            //   768 B   => 59776 B total

    const int lane = threadIdx.x & 31;
    const int wave = threadIdx.x >> 5;
    const int b    = blockIdx.x * WPG + wave;
    const int t    = threadIdx.x;

    // ---- stage Ss^T and padded Q into LDS (coalesced global reads)
    for (int i = t; i < NRATES * NSPEC; i += 32 * WPG) {
        int row = i >> 5, col = i & 31;          // Ss[row][col]
        lds_SsT[col * SST_STR + row] = SsG[i];
    }
    for (int i = t; i < NRATES * NRATES; i += 32 * WPG) {
        lds_Q[(i >> 6) * Q_STR + (i & 63)] = Qg[i];
    }
    __syncthreads();

    float* Mw  = &lds_M[wave][0];
    float* vec = &lds_vec[wave][0];

    // ---- per-sample inputs
    const float mc = maxc[lane];                       // lane in 0..31
    const float dc = dc_in[b * NSPEC + lane] * mc;
    const float cp = c_in [b * NSPEC + lane] * mc;

    // p = -2 * dc @ Ss^T  -> p[j] = -2 * sum_k dc[k]*SsT[k][j]
    vec[lane] = dc;
    LDS_SYNC();
    float p0 = 0.f, p1 = 0.f;
    for (int k = 0; k < NSPEC; ++k) {
        float d = vec[k];
        p0 += lds_SsT[k * SST_STR + lane]      * d;
        p1 += lds_SsT[k * SST_STR + lane + 32] * d;
    }
    p0 *= -2.f; p1 *= -2.f;

    // h = [c_prev ; 0], state x=0, s=1, z=1
    const float h0 = cp;
    float x0 = 0.f, x1 = 0.f;
    float s0 = 1.f, s1 = 1.f, s2 = 1.f;
    float z0 = 1.f, z1 = 1.f, z2 = 1.f;

    const int half = lane >> 4;   // 0: lanes 0-15, 1: lanes 16-31
    const int lan  = lane & 15;

#pragma unroll 1
    for (int it = 0; it < NITER; ++it) {
        // ---- broadcast x
        LDS_SYNC();
        vec[lane] = x0; vec[lane + 32] = x1;
        LDS_SYNC();

        // Ss^T x  (lane owns species 'lane')
        float stx = 0.f;
        for (int r = 0; r < NRATES; ++r)
            stx += lds_SsT[lane * SST_STR + r] * vec[r];

        // Q x  (from LDS, conflict-free)
        float q0 = 0.f, q1 = 0.f;
        for (int k = 0; k < NRATES; ++k) {
            float xk = vec[k];
            q0 += lds_Q[k * Q_STR + lane]      * xk;
            q1 += lds_Q[k * Q_STR + lane + 32] * xk;
        }

        // r_p = G x + s - h ;  G x = [-Ss^T x ; -x]
        float rp0 = -stx + s0 - h0;
        float rp1 = -x0  + s1;
        float rp2 = -x1  + s2;

        // ---- broadcast z_top ; G^T z = -Ss z_top - z_bot
        LDS_SYNC();
        vec[lane] = z0;
        LDS_SYNC();
        float sz0 = 0.f, sz1 = 0.f;
        for (int k = 0; k < NSPEC; ++k) {
            float zk = vec[k];
            sz0 += lds_SsT[k * SST_STR + lane]      * zk;
            sz1 += lds_SsT[k * SST_STR + lane + 32] * zk;
        }
        float rd0 = q0 + p0 - sz0 - z1;
        float rd1 = q1 + p1 - sz1 - z2;

        // ---- mu, complementarity residual, scaling
        float mu = wave_sum32(s0 * z0 + s1 * z1 + s2 * z2) * (1.f / NCON);
        float sm  = SIGMA * mu;
        float rc0 = s0 * z0 - sm, rc1 = s1 * z1 - sm, rc2 = s2 * z2 - sm;
        float w0  = z0 / s0,      w1  = z1 / s1,      w2  = z2 / s2;
        float u0  = (z0 * rp0 - rc0) / s0;
        float u1  = (z1 * rp1 - rc1) / s1;
        float u2  = (z2 * rp2 - rc2) / s2;

        // rhs = -r_d - G^T u = -r_d + Ss u_top + u_bot
        LDS_SYNC();
        vec[lane] = u0;
        LDS_SYNC();
        float su0 = 0.f, su1 = 0.f;
        for (int k = 0; k < NSPEC; ++k) {
            float uk = vec[k];
            su0 += lds_SsT[k * SST_STR + lane]      * uk;
            su1 += lds_SsT[k * SST_STR + lane + 32] * uk;
        }
        float rhs0 = -rd0 + su0 + u1;
        float rhs1 = -rd1 + su1 + u2;

        // ---- broadcast w_top only (w_bot added from registers afterwards)
        LDS_SYNC();
        vec[lane] = w0;
        LDS_SYNC();

        // ================= M = Q + Ss diag(w_top) Ss^T  (WMMA, K=32) ========
        // Row of 4 tiles per I: A-frag shared, 4 independent accumulators.
#pragma unroll 1
        for (int I = 0; I < 4; ++I) {
            const int arow = I * 16 + lan;
            v8f c0, c1, c2, c3;
#pragma unroll
            for (int v = 0; v < 8; ++v) {   // C init: Q tile row
                int qoff = (I * 16 + v + half * 8) * Q_STR + lan;
                c0[v] = lds_Q[qoff];
                c1[v] = lds_Q[qoff + 16];
                c2[v] = lds_Q[qoff + 32];
                c3[v] = lds_Q[qoff + 48];
            }
#pragma unroll 2
            for (int ks = 0; ks < 8; ++ks) {
                const int k = ks * 4 + half * 2;
                v2f a, b0, b1, b2, b3;
                a[0] = lds_SsT[ k      * SST_STR + arow] * vec[k];
                a[1] = lds_SsT[(k + 1) * SST_STR + arow] * vec[k + 1];
                const int boff = k * SST_STR + lan;
                b0[0] = lds_SsT[boff];           b0[1] = lds_SsT[boff + SST_STR];
                b1[0] = lds_SsT[boff + 16];      b1[1] = lds_SsT[boff + SST_STR + 16];
                b2[0] = lds_SsT[boff + 32];      b2[1] = lds_SsT[boff + SST_STR + 32];
                b3[0] = lds_SsT[boff + 48];      b3[1] = lds_SsT[boff + SST_STR + 48];
                c0 = __builtin_amdgcn_wmma_f32_16x16x4_f32(false, a, false, b0, (short)0, c0, false, false);
                c1 = __builtin_amdgcn_wmma_f32_16x16x4_f32(false, a, false, b1, (short)0, c1, false, false);
                c2 = __builtin_amdgcn_wmma_f32_16x16x4_f32(false, a, false, b2, (short)0, c2, false, false);
                c3 = __builtin_amdgcn_wmma_f32_16x16x4_f32(false, a, false, b3, (short)0, c3, false, false);
            }
#pragma unroll
            for (int v = 0; v < 8; ++v) {
                int moff = (I * 16 + v + half * 8) * M_STR + lan;
                Mw[moff]      = c0[v];
                Mw[moff + 16] = c1[v];
                Mw[moff + 32] = c2[v];
                Mw[moff + 48] = c3[v];
            }
        }
        LDS_SYNC();
        // diagonal: + eps + w_bot  (w_bot values live in this lane's registers)
        Mw[ lane       * M_STR + lane]      += REG_JIT + w1;
        Mw[(lane + 32) * M_STR + lane + 32] += REG_JIT + w2;
        LDS_SYNC();

        // ================= blocked Cholesky, NB=16 ==========================
#pragma unroll 1
        for (int kb = 0; kb < 4; ++kb) {
            const int j0 = kb * 16;
            // ---- panel factor: columns j0..j0+15, full height
#pragma unroll 1
            for (int jj = 0; jj < 16; ++jj) {
                const int j = j0 + jj;
                float ljj = sqrtf(Mw[j * M_STR + j]);
                float inv = 1.f / ljj;
                if (lane == 0) Mw[j * M_STR + j] = ljj;
                int r = j + 1 + lane;
                if (r < NRATES)      Mw[ r       * M_STR + j] *= inv;
                if (r + 32 < NRATES) Mw[(r + 32) * M_STR + j] *= inv;
                LDS_SYNC();
                // trailing update restricted to remaining columns of the block
#pragma unroll 1
                for (int cc = j + 1; cc < j0 + 16; ++cc) {
                    float ljc = Mw[cc * M_STR + j];
                    int rr = cc + lane;
                    if (rr < NRATES)
                        Mw[ rr       * M_STR + cc] -= Mw[ rr       * M_STR + j] * ljc;
                    if (rr + 32 < NRATES)
                        Mw[(rr + 32) * M_STR + cc] -= Mw[(rr + 32) * M_STR + j] * ljc;
                }
                LDS_SYNC();
            }
            // ---- trailing SYRK on WMMA: A22 -= P P^T (lower-triangle tiles)
            const int nt = 3 - kb;          // remaining 16-row blocks
#pragma unroll 1
            for (int ti = 0; ti < nt; ++ti) {
                const int rowbase = (kb + 1 + ti) * 16;
#pragma unroll 1
                for (int tj = 0; tj <= ti; ++tj) {
                    const int colbase = (kb + 1 + tj) * 16;
                    v8f c;
#pragma unroll
                    for (int v = 0; v < 8; ++v)
                        c[v] = Mw[(rowbase + v + half * 8) * M_STR + colbase + lan];
#pragma unroll
                    for (int ks = 0; ks < 4; ++ks) {
                        const int k = j0 + ks * 4 + half * 2;
                        v2f a, bq;
                        // A = -P[rowblock]  (fold the subtraction into A)
                        a[0]  = -Mw[(rowbase + lan) * M_STR + k];
                        a[1]  = -Mw[(rowbase + lan) * M_STR + k + 1];
                        // B = P[colblock]^T
                        bq[0] =  Mw[(colbase + lan) * M_STR + k];
                        bq[1] =  Mw[(colbase + lan) * M_STR + k + 1];
                        c = __builtin_amdgcn_wmma_f32_16x16x4_f32(
                                false, a, false, bq, (short)0, c, false, false);
                    }
#pragma unroll
                    for (int v = 0; v < 8; ++v)
                        Mw[(rowbase + v + half * 8) * M_STR + colbase + lan] = c[v];
                }
            }
            LDS_SYNC();
        }

        // ================= triangular solves ================================
        // forward solve L y = rhs
        vec[lane] = rhs0; vec[lane + 32] = rhs1;
        LDS_SYNC();
#pragma unroll 1
        for (int j = 0; j < NRATES; ++j) {
            float yj = vec[j] / Mw[j * M_STR + j];
            if (lane == 0) vec[j] = yj;
            int r = j + 1 + lane;
            if (r < NRATES)      vec[r]      -= Mw[ r       * M_STR + j] * yj;
            if (r + 32 < NRATES) vec[r + 32] -= Mw[(r + 32) * M_STR + j] * yj;
            LDS_SYNC();
        }
        // backward solve L^T dx = y   (L^T[r][j] = L[j][r])
#pragma unroll 1
        for (int j = NRATES - 1; j >= 0; --j) {
            float dj = vec[j] / Mw[j * M_STR + j];
            if (lane == 0) vec[j] = dj;
            if (lane < j)      vec[lane]      -= Mw[j * M_STR + lane]      * dj;
            if (lane + 32 < j) vec[lane + 32] -= Mw[j * M_STR + lane + 32] * dj;
            LDS_SYNC();
        }
        LDS_SYNC();
        float dx0 = vec[lane], dx1 = vec[lane + 32];

        // ---- ds = -r_p - G dx ;  G dx = [-Ss^T dx ; -dx]
        float stdx = 0.f;
        for (int r = 0; r < NRATES; ++r)
            stdx += lds_SsT[lane * SST_STR + r] * vec[r];
        float ds0 = -rp0 + stdx;
        float ds1 = -rp1 + dx0;
        float ds2 = -rp2 + dx1;
        float dz0 = (-rc0 - z0 * ds0) / s0;
        float dz1 = (-rc1 - z1 * ds1) / s1;
        float dz2 = (-rc2 - z2 * ds2) / s2;

        // ---- fraction-to-boundary step length
        const float INF = __builtin_inff();
        float a = INF;
        a = fminf(a, ds0 < 0.f ? -s0 / ds0 : INF);
        a = fminf(a, ds1 < 0.f ? -s1 / ds1 : INF);
        a = fminf(a, ds2 < 0.f ? -s2 / ds2 : INF);
        a = fminf(a, dz0 < 0.f ? -z0 / dz0 : INF);
        a = fminf(a, dz1 < 0.f ? -z1 / dz1 : INF);
        a = fminf(a, dz2 < 0.f ? -z2 / dz2 : INF);
        a = wave_min32(a);
        float alpha = fminf(1.f, TAU * a);

        x0 += alpha * dx0; x1 += alpha * dx1;
        s0 += alpha * ds0; s1 += alpha * ds1; s2 += alpha * ds2;
        z0 += alpha * dz0; z1 += alpha * dz1; z2 += alpha * dz2;
        LDS_SYNC();
    }

    // ---- output: out[b][sp] = (sum_r x[r]*Ss[r][sp]) / max_c[sp]
    vec[lane] = x0; vec[lane + 32] = x1;
    LDS_SYNC();
    float o = 0.f;
    for (int r = 0; r < NRATES; ++r)
        o += lds_SsT[lane * SST_STR + r] * vec[r];
    out[b * NSPEC + lane] = o / mc;
}

extern "C" void kernel_launch(void* const* d_in, const int* in_sizes, int n_in,
                              void* d_out, int out_size, void* d_ws, size_t ws_size,
                              hipStream_t stream) {
    const float* dc_in = (const float*)d_in[0];   // [B,32]
    const float* c_in  = (const float*)d_in[1];   // [B,32]
    const float* Ss    = (const float*)d_in[2];   // [64,32]
    const float* Q     = (const float*)d_in[3];   // [64,64]
    // d_in[4] = G (96x64) — unused: G == [-Ss^T; -I] by construction
    const float* maxc  = (const float*)d_in[5];   // [32]
    float* out = (float*)d_out;                   // [B,32]

    const int B = in_sizes[0] / NSPEC;            // 4096
    dim3 grid((B + WPG - 1) / WPG);
    dim3 block(32 * WPG);
    hipLaunchKernelGGL(optlayer_ipm_kernel, grid, block, 0, stream,
                       dc_in, c_in, Ss, Q, maxc, out);
}